// MambaClassifier_76914274337257
// MI455X (gfx1250) — compile-verified
//
#include <hip/hip_runtime.h>
#include <hip/hip_bf16.h>
#include <math.h>

// ---------------- problem constants ----------------
#define B_    2
#define L_    1024
#define DM    1024          // D_MODEL
#define DI    2048          // D_INNER
#define DST   16            // D_STATE
#define DCONV 4
#define DTR   64            // DT_RANK
#define NCLS  10
#define XW    96            // DT_RANK + 2*D_STATE
#define EPS_  1e-5f
#define MTOK  (B_ * L_)     // 2048 token rows

typedef __attribute__((ext_vector_type(2))) float v2f;
typedef __attribute__((ext_vector_type(8))) float v8f;
typedef __attribute__((ext_vector_type(4))) int   v4i;

#define AS1 __attribute__((address_space(1)))
#define AS3 __attribute__((address_space(3)))

// ---- gfx1250 async global->LDS path (guarded; falls back to reg staging) ----
#if defined(__has_builtin)
#if __has_builtin(__builtin_amdgcn_global_load_async_to_lds_b128)
#define HAVE_ASYNC_LDS 1
#endif
#endif
#ifndef HAVE_ASYNC_LDS
#define HAVE_ASYNC_LDS 0
#endif

#if HAVE_ASYNC_LDS
__device__ __forceinline__ void async_lds_b128(const float* gsrc, float* ldst) {
    // signature (probe-confirmed): (v4i AS1* src, v4i AS3* dst, imm offset, imm cpol)
    __builtin_amdgcn_global_load_async_to_lds_b128(
        (AS1 v4i*)gsrc, (AS3 v4i*)ldst, 0, 0);
}
__device__ __forceinline__ void wait_async0() {
#if __has_builtin(__builtin_amdgcn_s_wait_asynccnt)
    __builtin_amdgcn_s_wait_asynccnt(0);
#else
    asm volatile("s_wait_asynccnt 0x0" ::: "memory");
#endif
}
#endif

// =====================================================================
// FP32 WMMA GEMM:  C[M,N] (+)= A[M,K](lda) * W[N,K]^T   (W row-major N x K)
//
// Block: 256 threads = 8 waves. Output tile 128 x (16*NB).
// K processed in KC=32 chunks, double-buffered through LDS.
//  - ASYNC path: GLOBAL_LOAD_ASYNC_TO_LDS_B128 writes chunk c+1 straight
//    into the alternate LDS buffer while WMMA consumes chunk c;
//    s_wait_asynccnt 0 + ONE barrier per chunk.
//  - Fallback: coalesced b128 global loads -> regs -> ds_store (2 barriers).
//
// V_WMMA_F32_16X16X4_F32 fragment layout (ISA 7.12.2):
//   A (lane): row = lane&15, holds K = {koff, koff+1}, koff = 2*(lane>=16)
//   B (lane): col = lane&15, same K addressing (W row-major N x K)
//   C/D (lane, r): row = 8*(lane>=16) + r, col = lane&15
// =====================================================================
#define KC    32
#define SW    36     // LDS row stride (floats): 32 + 4 pad

template <int NB, int EPI>   // EPI: 0 = store, 1 = accumulate (+=)
__global__ __launch_bounds__(256)
void gemm_wmma_f32(const float* __restrict__ A, int lda,
                   const float* __restrict__ W,
                   float* __restrict__ C,
                   int M, int N, int K)
{
    constexpr int NT = 16 * NB;       // N tile per block
    constexpr int BP = NT / 32;       // W-panel cooperative load passes

    __shared__ float sA[2][128 * SW];
    __shared__ float sB[2][NT * SW];

    const int tid  = threadIdx.x;
    const int lane = tid & 31;
    const int wave = tid >> 5;
    const int mblk = blockIdx.x * 128;
    const int n0   = blockIdx.y * NT;

    // WMMA fragment addressing
    const int fm   = wave * 16 + (lane & 15);   // sA row
    const int fn   = (lane & 15);               // sB base row
    const int koff = ((lane >> 4) << 1);        // 0 or 2

    // cooperative-load mapping: 8 threads per row (float4 / b128 each)
    const int lr = tid >> 3;          // 0..31
    const int lc = (tid & 7) * 4;     // 0,4,...,28

    v8f acc[NB];
#pragma unroll
    for (int i = 0; i < NB; ++i) acc[i] = v8f{0,0,0,0,0,0,0,0};

    const int nChunks = K / KC;

#if HAVE_ASYNC_LDS
    // ---------------- async global->LDS pipeline ----------------
    // prologue: chunk 0 -> buf 0
#pragma unroll
    for (int p = 0; p < 4; ++p)
        async_lds_b128(A + (size_t)(mblk + lr + 32 * p) * lda + lc,
                       &sA[0][(lr + 32 * p) * SW + lc]);
#pragma unroll
    for (int p = 0; p < BP; ++p)
        async_lds_b128(W + (size_t)(n0 + lr + 32 * p) * K + lc,
                       &sB[0][(lr + 32 * p) * SW + lc]);
    wait_async0();
    __syncthreads();

    for (int c = 0; c < nChunks; ++c) {
        const int pb = c & 1;

        if (c + 1 < nChunks) {   // issue chunk c+1 straight into buf pb^1
            const int kc = (c + 1) * KC;
#pragma unroll
            for (int p = 0; p < 4; ++p)
                async_lds_b128(A + (size_t)(mblk + lr + 32 * p) * lda + kc + lc,
                               &sA[pb ^ 1][(lr + 32 * p) * SW + lc]);
#pragma unroll
            for (int p = 0; p < BP; ++p)
                async_lds_b128(W + (size_t)(n0 + lr + 32 * p) * K + kc + lc,
                               &sB[pb ^ 1][(lr + 32 * p) * SW + lc]);
        }

        // WMMA compute on chunk c from LDS
#pragma unroll
        for (int k = 0; k < KC; k += 4) {
            v2f a = *(const v2f*)&sA[pb][fm * SW + k + koff];
#pragma unroll
            for (int i = 0; i < NB; ++i) {
                v2f b = *(const v2f*)&sB[pb][(fn + 16 * i) * SW + k + koff];
                acc[i] = __builtin_amdgcn_wmma_f32_16x16x4_f32(
                    false, a, false, b, (short)0, acc[i], false, false);
            }
        }

        if (c + 1 < nChunks) {
            wait_async0();       // my async writes for chunk c+1 landed
            __syncthreads();     // everyone done reading pb, writes visible
        }
    }
#else
    // ---------------- register-staged fallback pipeline ----------------
    float4 ra[4];
    float4 rb[BP];

#pragma unroll
    for (int p = 0; p < 4; ++p)
        ra[p] = *(const float4*)(A + (size_t)(mblk + lr + 32 * p) * lda + lc);
#pragma unroll
    for (int p = 0; p < BP; ++p)
        rb[p] = *(const float4*)(W + (size_t)(n0 + lr + 32 * p) * K + lc);
#pragma unroll
    for (int p = 0; p < 4; ++p)
        *(float4*)&sA[0][(lr + 32 * p) * SW + lc] = ra[p];
#pragma unroll
    for (int p = 0; p < BP; ++p)
        *(float4*)&sB[0][(lr + 32 * p) * SW + lc] = rb[p];
    __syncthreads();

    for (int c = 0; c < nChunks; ++c) {
        const int pb = c & 1;

        if (c + 1 < nChunks) {
            const int kc = (c + 1) * KC;
#pragma unroll
            for (int p = 0; p < 4; ++p)
                ra[p] = *(const float4*)(A + (size_t)(mblk + lr + 32 * p) * lda + kc + lc);
#pragma unroll
            for (int p = 0; p < BP; ++p)
                rb[p] = *(const float4*)(W + (size_t)(n0 + lr + 32 * p) * K + kc + lc);
        }

#pragma unroll
        for (int k = 0; k < KC; k += 4) {
            v2f a = *(const v2f*)&sA[pb][fm * SW + k + koff];
#pragma unroll
            for (int i = 0; i < NB; ++i) {
                v2f b = *(const v2f*)&sB[pb][(fn + 16 * i) * SW + k + koff];
                acc[i] = __builtin_amdgcn_wmma_f32_16x16x4_f32(
                    false, a, false, b, (short)0, acc[i], false, false);
            }
        }
        __syncthreads();

        if (c + 1 < nChunks) {
#pragma unroll
            for (int p = 0; p < 4; ++p)
                *(float4*)&sA[pb ^ 1][(lr + 32 * p) * SW + lc] = ra[p];
#pragma unroll
            for (int p = 0; p < BP; ++p)
                *(float4*)&sB[pb ^ 1][(lr + 32 * p) * SW + lc] = rb[p];
            __syncthreads();
        }
    }
#endif

    // ---- epilogue ----
    const int crow = mblk + wave * 16 + ((lane >> 4) << 3);
    const int ccol = n0 + (lane & 15);
#pragma unroll
    for (int i = 0; i < NB; ++i) {
#pragma unroll
        for (int r = 0; r < 8; ++r) {
            size_t idx = (size_t)(crow + r) * N + (ccol + 16 * i);
            if (EPI == 0) C[idx] = acc[i][r];
            else          C[idx] += acc[i][r];
        }
    }
}

// =====================================================================
// RMSNorm: one block (256 thr) per token row of width DM
// =====================================================================
__global__ __launch_bounds__(256)
void rmsnorm_kernel(const float* __restrict__ x, const float* __restrict__ w,
                    float* __restrict__ y)
{
    const int row = blockIdx.x;
    const int tid = threadIdx.x;
    const float* xr = x + (size_t)row * DM;
    float* yr = y + (size_t)row * DM;

    __shared__ float red[256];
    float s = 0.f;
    for (int d = tid; d < DM; d += 256) { float v = xr[d]; s += v * v; }
    red[tid] = s; __syncthreads();
    for (int off = 128; off > 0; off >>= 1) {
        if (tid < off) red[tid] += red[tid + off];
        __syncthreads();
    }
    const float scale = rsqrtf(red[0] / (float)DM + EPS_);
    for (int d = tid; d < DM; d += 256) yr[d] = xr[d] * scale * w[d];
}

// =====================================================================
// Depthwise causal conv (width 4) + bias + SiLU; rolling register window.
// =====================================================================
__global__ __launch_bounds__(256)
void conv_silu_kernel(const float* __restrict__ xz,
                      const float* __restrict__ cw,  // (DI, 4)
                      const float* __restrict__ cb,  // (DI)
                      float* __restrict__ xc)        // (B*L, DI)
{
    const int b = blockIdx.y;
    const int d = blockIdx.x * blockDim.x + threadIdx.x;
    const float w0 = cw[d * 4 + 0], w1 = cw[d * 4 + 1];
    const float w2 = cw[d * 4 + 2], w3 = cw[d * 4 + 3];
    const float bias = cb[d];

    float xm3 = 0.f, xm2 = 0.f, xm1 = 0.f;
    for (int t = 0; t < L_; ++t) {
        const size_t row = (size_t)b * L_ + t;
        float x0 = xz[row * (2 * DI) + d];
        float acc = w0 * xm3 + w1 * xm2 + w2 * xm1 + w3 * x0 + bias;
        float sg = 1.f / (1.f + expf(-acc));
        xc[row * DI + d] = acc * sg;
        xm3 = xm2; xm2 = xm1; xm1 = x0;
    }
}

// =====================================================================
// delta = softplus(dt_raw + dtb)
// =====================================================================
__global__ __launch_bounds__(256)
void bias_softplus_kernel(float* __restrict__ delta,
                          const float* __restrict__ dtb)
{
    const int idx = blockIdx.x * blockDim.x + threadIdx.x;
    const int d = idx & (DI - 1);
    float v = delta[idx] + dtb[d];
    delta[idx] = (v > 20.f) ? v : log1pf(expf(v));
}

// =====================================================================
// Selective scan, fused D-term + SiLU gate.
// One thread per (b, d), 16 states in registers.
// B_t/C_t broadcast via LDS in chunks of 8 timesteps (2 barriers / 8 t).
// =====================================================================
__global__ __launch_bounds__(256)
void scan_kernel(const float* __restrict__ xc,     // (B*L, DI)  u
                 const float* __restrict__ delta,  // (B*L, DI)
                 const float* __restrict__ A_log,  // (DI, DST)
                 const float* __restrict__ xdbl,   // (B*L, 96): [64:80)=B, [80:96)=C
                 const float* __restrict__ Dp,     // (DI)
                 const float* __restrict__ xz,     // res at col DI+d, row width 2*DI
                 float* __restrict__ yg)           // (B*L, DI)
{
    const int b = blockIdx.y;
    const int d = blockIdx.x * blockDim.x + threadIdx.x;
    const int tid = threadIdx.x;

    float Av[DST];
#pragma unroll
    for (int n = 0; n < DST; ++n) Av[n] = -expf(A_log[(size_t)d * DST + n]);
    const float Dd = Dp[d];

    float h[DST];
#pragma unroll
    for (int n = 0; n < DST; ++n) h[n] = 0.f;

    __shared__ float sBC[8 * 32];    // 8 timesteps x (16 B + 16 C)

    for (int t0 = 0; t0 < L_; t0 += 8) {
        {
            const int tt = tid >> 5, j = tid & 31;
            sBC[tid] = xdbl[((size_t)b * L_ + t0 + tt) * XW + DTR + j];
        }
        __syncthreads();

#pragma unroll
        for (int tt = 0; tt < 8; ++tt) {
            const size_t row = (size_t)b * L_ + t0 + tt;
            const float* bc = &sBC[tt * 32];

            const float u  = xc[row * DI + d];
            const float dt = delta[row * DI + d];
            const float du = dt * u;
            float y = 0.f;
#pragma unroll
            for (int n = 0; n < DST; ++n) {
                float dA = expf(dt * Av[n]);
                h[n] = dA * h[n] + du * bc[n];
                y += h[n] * bc[16 + n];
            }
            y += u * Dd;

            const float res = xz[row * (2 * DI) + DI + d];
            const float sg = 1.f / (1.f + expf(-res));
            yg[row * DI + d] = y * (res * sg);
        }
        __syncthreads();
    }
}

// =====================================================================
// Head: final rmsnorm on last token + fc. One block per batch element.
// =====================================================================
__global__ __launch_bounds__(256)
void head_kernel(const float* __restrict__ h,
                 const float* __restrict__ nw,
                 const float* __restrict__ fcw, const float* __restrict__ fcb,
                 float* __restrict__ out)
{
    const int b = blockIdx.x;
    const int tid = threadIdx.x;
    const float* hv = h + ((size_t)b * L_ + (L_ - 1)) * DM;

    __shared__ float red[256];
    __shared__ float scale;

    float s = 0.f;
    for (int d = tid; d < DM; d += 256) { float v = hv[d]; s += v * v; }
    red[tid] = s; __syncthreads();
    for (int off = 128; off > 0; off >>= 1) {
        if (tid < off) red[tid] += red[tid + off];
        __syncthreads();
    }
    if (tid == 0) scale = rsqrtf(red[0] / (float)DM + EPS_);
    __syncthreads();

    for (int c = 0; c < NCLS; ++c) {
        float p = 0.f;
        for (int d = tid; d < DM; d += 256)
            p += hv[d] * scale * nw[d] * fcw[(size_t)c * DM + d];
        red[tid] = p; __syncthreads();
        for (int off = 128; off > 0; off >>= 1) {
            if (tid < off) red[tid] += red[tid + off];
            __syncthreads();
        }
        if (tid == 0) out[b * NCLS + c] = red[0] + fcb[c];
        __syncthreads();
    }
}

// =====================================================================
// host launcher
// =====================================================================
extern "C" void kernel_launch(void* const* d_in, const int* in_sizes, int n_in,
                              void* d_out, int out_size, void* d_ws, size_t ws_size,
                              hipStream_t stream)
{
    const float* x        = (const float*)d_in[0];   // (2,1024,1024)
    const float* norm_w   = (const float*)d_in[1];   // (2,1024)
    const float* in_w     = (const float*)d_in[2];   // (2,4096,1024)
    const float* conv_w   = (const float*)d_in[3];   // (2,2048,4)
    const float* conv_b   = (const float*)d_in[4];   // (2,2048)
    const float* xp_w     = (const float*)d_in[5];   // (2,96,2048)
    const float* dt_w     = (const float*)d_in[6];   // (2,2048,64)
    const float* dt_b     = (const float*)d_in[7];   // (2,2048)
    const float* A_log    = (const float*)d_in[8];   // (2,2048,16)
    const float* Dp       = (const float*)d_in[9];   // (2,2048)
    const float* out_w    = (const float*)d_in[10];  // (2,1024,2048)
    const float* norm_f_w = (const float*)d_in[11];  // (1024)
    const float* fc_w     = (const float*)d_in[12];  // (10,1024)
    const float* fc_b     = (const float*)d_in[13];  // (10)
    float* out = (float*)d_out;

    // ---- workspace layout (floats) ----
    float* ws   = (float*)d_ws;
    float* h    = ws;                                // MTOK*DM
    float* xn   = h    + (size_t)MTOK * DM;          // MTOK*DM
    float* xz   = xn   + (size_t)MTOK * DM;          // MTOK*2*DI
    float* xc   = xz   + (size_t)MTOK * 2 * DI;      // MTOK*DI
    float* xdbl = xc   + (size_t)MTOK * DI;          // MTOK*96
    float* dlt  = xdbl + (size_t)MTOK * XW;          // MTOK*DI
    float* yg   = dlt  + (size_t)MTOK * DI;          // MTOK*DI

    // residual stream h := x  (fresh every call; graph-capture safe)
    (void)hipMemcpyAsync(h, x, (size_t)MTOK * DM * sizeof(float),
                         hipMemcpyDeviceToDevice, stream);

    for (int i = 0; i < 2; ++i) {
        const float* in_w_i  = in_w  + (size_t)i * 2 * DI * DM;
        const float* xp_w_i  = xp_w  + (size_t)i * XW * DI;
        const float* dt_w_i  = dt_w  + (size_t)i * DI * DTR;
        const float* out_w_i = out_w + (size_t)i * DM * DI;

        // 1) rmsnorm
        rmsnorm_kernel<<<MTOK, 256, 0, stream>>>(h, norm_w + (size_t)i * DM, xn);

        // 2) in_proj: xz = xn @ in_w^T   (M=2048, N=4096, K=1024)
        gemm_wmma_f32<4, 0><<<dim3(MTOK / 128, (2 * DI) / 64), 256, 0, stream>>>(
            xn, DM, in_w_i, xz, MTOK, 2 * DI, DM);

        // 3) depthwise conv + SiLU
        conv_silu_kernel<<<dim3(DI / 256, B_), 256, 0, stream>>>(
            xz, conv_w + (size_t)i * DI * DCONV, conv_b + (size_t)i * DI, xc);

        // 4) x_proj: xdbl = xc @ xp_w^T  (M=2048, N=96, K=2048)
        gemm_wmma_f32<2, 0><<<dim3(MTOK / 128, XW / 32), 256, 0, stream>>>(
            xc, DI, xp_w_i, xdbl, MTOK, XW, DI);

        // 5) dt_proj: dlt = xdbl[:, :64] @ dt_w^T  (M=2048, N=2048, K=64, lda=96)
        gemm_wmma_f32<4, 0><<<dim3(MTOK / 128, DI / 64), 256, 0, stream>>>(
            xdbl, XW, dt_w_i, dlt, MTOK, DI, DTR);

        // 6) delta = softplus(dlt + dt_b)
        bias_softplus_kernel<<<(MTOK * DI) / 256, 256, 0, stream>>>(
            dlt, dt_b + (size_t)i * DI);

        // 7) selective scan + D term + SiLU gating
        scan_kernel<<<dim3(DI / 256, B_), 256, 0, stream>>>(
            xc, dlt, A_log + (size_t)i * DI * DST, xdbl,
            Dp + (size_t)i * DI, xz, yg);

        // 8) out_proj with residual accumulate: h += yg @ out_w^T
        gemm_wmma_f32<4, 1><<<dim3(MTOK / 128, DM / 64), 256, 0, stream>>>(
            yg, DI, out_w_i, h, MTOK, DM, DI);
    }

    // final rmsnorm (last token only) + fc
    head_kernel<<<B_, 256, 0, stream>>>(h, norm_f_w, fc_w, fc_b, out);
}